// Encoder_28939489640686
// MI455X (gfx1250) — compile-verified
//
#include <hip/hip_runtime.h>

typedef __attribute__((ext_vector_type(2))) float v2f;
typedef __attribute__((ext_vector_type(8))) float v8f;

#define NNODES 200000
#define NEDGES 800000

// ---------------- degree / norm ----------------
__global__ void k_deg(const long long* __restrict__ dst, float* __restrict__ deg, int E) {
    int e = blockIdx.x * blockDim.x + threadIdx.x;
    if (e < E) atomicAdd(&deg[(int)dst[e]], 1.0f);
}

__global__ void k_dinv(const float* __restrict__ deg, float* __restrict__ dinv, int n) {
    int i = blockIdx.x * blockDim.x + threadIdx.x;
    if (i < n) dinv[i] = rsqrtf(deg[i] + 1.0f);   // +1 = self loop; always > 0
}

__global__ void k_norm(const long long* __restrict__ src, const long long* __restrict__ dst,
                       const float* __restrict__ dinv, float* __restrict__ nrm, int E) {
    int e = blockIdx.x * blockDim.x + threadIdx.x;
    if (e < E) nrm[e] = dinv[(int)src[e]] * dinv[(int)dst[e]];
}

// ---------------- packing (zero pad) ----------------
__global__ void k_pack_x(const float* __restrict__ x, float* __restrict__ xp,
                         int n, int F, int Fp) {
    int i = blockIdx.x * blockDim.x + threadIdx.x;
    if (i >= n * Fp) return;
    int node = i / Fp, f = i % Fp;
    xp[i] = (f < F) ? x[(size_t)node * F + f] : 0.0f;
}

// Wt[n*Kp + k] = W[k*Nin + n], zero padded to [Np x Kp]
__global__ void k_pack_wt(const float* __restrict__ W, float* __restrict__ Wt,
                          int Kin, int Kp, int Nin, int Np) {
    int i = blockIdx.x * blockDim.x + threadIdx.x;
    if (i >= Np * Kp) return;
    int nn = i / Kp, k = i % Kp;
    Wt[i] = (k < Kin && nn < Nin) ? W[(size_t)k * Nin + nn] : 0.0f;
}

// ---------------- fp32 WMMA GEMM: H[M x Np] = A[M x Kp] @ W ----------------
// One wave computes a 32-row x 80-col strip (2x5 = 10 16x16 tiles).
// K stepped by 4; manual 2x unroll with ping-pong fragment sets so no
// cross-iteration register rotation is needed and loads stay in flight
// across each 10-WMMA burst.

struct Frag {
    v2f a0, a1, b0, b1, b2, b3, b4;
};

__device__ __forceinline__ Frag load_frag(const float* arow0, const float* arow1,
                                          const float* bcol, size_t ts, int ko) {
    Frag f;
    f.a0 = *(const v2f*)(arow0 + ko);
    f.a1 = *(const v2f*)(arow1 + ko);
    f.b0 = *(const v2f*)(bcol + ko);
    f.b1 = *(const v2f*)(bcol + ts     + ko);
    f.b2 = *(const v2f*)(bcol + ts * 2 + ko);
    f.b3 = *(const v2f*)(bcol + ts * 3 + ko);
    f.b4 = *(const v2f*)(bcol + ts * 4 + ko);
    return f;
}

__device__ __forceinline__ void mma_step(const Frag& f,
    v8f& c00, v8f& c01, v8f& c02, v8f& c03, v8f& c04,
    v8f& c10, v8f& c11, v8f& c12, v8f& c13, v8f& c14) {
    c00 = __builtin_amdgcn_wmma_f32_16x16x4_f32(false, f.a0, false, f.b0, (short)0, c00, false, false);
    c10 = __builtin_amdgcn_wmma_f32_16x16x4_f32(false, f.a1, false, f.b0, (short)0, c10, false, false);
    c01 = __builtin_amdgcn_wmma_f32_16x16x4_f32(false, f.a0, false, f.b1, (short)0, c01, false, false);
    c11 = __builtin_amdgcn_wmma_f32_16x16x4_f32(false, f.a1, false, f.b1, (short)0, c11, false, false);
    c02 = __builtin_amdgcn_wmma_f32_16x16x4_f32(false, f.a0, false, f.b2, (short)0, c02, false, false);
    c12 = __builtin_amdgcn_wmma_f32_16x16x4_f32(false, f.a1, false, f.b2, (short)0, c12, false, false);
    c03 = __builtin_amdgcn_wmma_f32_16x16x4_f32(false, f.a0, false, f.b3, (short)0, c03, false, false);
    c13 = __builtin_amdgcn_wmma_f32_16x16x4_f32(false, f.a1, false, f.b3, (short)0, c13, false, false);
    c04 = __builtin_amdgcn_wmma_f32_16x16x4_f32(false, f.a0, false, f.b4, (short)0, c04, false, false);
    c14 = __builtin_amdgcn_wmma_f32_16x16x4_f32(false, f.a1, false, f.b4, (short)0, c14, false, false);
}

__global__ __launch_bounds__(64) void k_gemm_wmma(
    const float* __restrict__ A,   // [M x Kp] row-major, Kp % 8 == 0 (even #k-steps)
    const float* __restrict__ Wt,  // [Np x Kp] = B transposed
    float* __restrict__ H,         // [M x Np]
    int Kp, int Np) {
    const int lane = threadIdx.x & 31;
    const int wave = threadIdx.x >> 5;
    const int m    = lane & 15;        // row within A tile / col within B,C tiles
    const int kh   = lane >> 4;        // 0|1 : k-half select
    const int row0 = (blockIdx.x * 2 + wave) * 32;
    const int col0 = blockIdx.y * 80;

    const float* arow0 = A  + (size_t)(row0 + m) * Kp;
    const float* arow1 = A  + (size_t)(row0 + 16 + m) * Kp;
    const float* bcol  = Wt + (size_t)(col0 + m) * Kp;
    const size_t ts = (size_t)16 * Kp;
    const int ko0 = 2 * kh;

    v8f c00 = {}, c01 = {}, c02 = {}, c03 = {}, c04 = {};
    v8f c10 = {}, c11 = {}, c12 = {}, c13 = {}, c14 = {};

    Frag fa = load_frag(arow0, arow1, bcol, ts, ko0);
    int k = 4;
    for (; k < Kp - 4; k += 8) {
        Frag fb = load_frag(arow0, arow1, bcol, ts, k + ko0);
        mma_step(fa, c00, c01, c02, c03, c04, c10, c11, c12, c13, c14);
        fa = load_frag(arow0, arow1, bcol, ts, k + 4 + ko0);
        mma_step(fb, c00, c01, c02, c03, c04, c10, c11, c12, c13, c14);
    }
    // k == Kp - 4 here (Kp/4 is even): one load + two computes remain
    Frag fb = load_frag(arow0, arow1, bcol, ts, k + ko0);
    mma_step(fa, c00, c01, c02, c03, c04, c10, c11, c12, c13, c14);
    mma_step(fb, c00, c01, c02, c03, c04, c10, c11, c12, c13, c14);

#pragma unroll
    for (int r = 0; r < 8; ++r) {
        float* p0 = H + (size_t)(row0 + r + 8 * kh) * Np + col0 + m;
        p0[0]  = c00[r];
        p0[16] = c01[r];
        p0[32] = c02[r];
        p0[48] = c03[r];
        p0[64] = c04[r];
        float* p1 = H + (size_t)(row0 + 16 + r + 8 * kh) * Np + col0 + m;
        p1[0]  = c10[r];
        p1[16] = c11[r];
        p1[32] = c12[r];
        p1[48] = c13[r];
        p1[64] = c14[r];
    }
}

// ---------------- AGG init: out = H * dinv^2 + bias (self-loop + bias) ----------------
__global__ void k_init_agg(const float* __restrict__ H, const float* __restrict__ dinv,
                           const float* __restrict__ bias, float* __restrict__ out,
                           int n, int Fp, int Fout, int outStride) {
    int q = Fp >> 2;
    int i = blockIdx.x * blockDim.x + threadIdx.x;
    if (i >= n * q) return;
    int node = i / q;
    int f = (i % q) << 2;
    if (f >= outStride) return;             // padded cols beyond out buffer (layer 3)
    float d = dinv[node];
    float d2 = d * d;
    float4 h = *(const float4*)(H + (size_t)node * Fp + f);
    float4 v;
    v.x = (f + 0 < Fout) ? h.x * d2 + bias[f + 0] : 0.0f;
    v.y = (f + 1 < Fout) ? h.y * d2 + bias[f + 1] : 0.0f;
    v.z = (f + 2 < Fout) ? h.z * d2 + bias[f + 2] : 0.0f;
    v.w = (f + 3 < Fout) ? h.w * d2 + bias[f + 3] : 0.0f;
    *(float4*)(out + (size_t)node * outStride + f) = v;
}

// ---------------- edge scatter: out[dst] += H[src] * norm ----------------
__global__ void k_scatter(const float* __restrict__ H, const long long* __restrict__ src,
                          const long long* __restrict__ dst, const float* __restrict__ nrm,
                          float* __restrict__ out, int E, int Fp, int Fout, int outStride,
                          int quarters) {
    long long i = (long long)blockIdx.x * blockDim.x + threadIdx.x;
    if (i >= (long long)E * quarters) return;
    int e = (int)(i / quarters);
    int f = (int)(i % quarters) << 2;
    float w = nrm[e];
    int s = (int)src[e];
    int d = (int)dst[e];
    float4 h = *(const float4*)(H + (size_t)s * Fp + f);
    float* o = out + (size_t)d * outStride + f;
    if (f + 0 < Fout) atomicAdd(o + 0, h.x * w);
    if (f + 1 < Fout) atomicAdd(o + 1, h.y * w);
    if (f + 2 < Fout) atomicAdd(o + 2, h.z * w);
    if (f + 3 < Fout) atomicAdd(o + 3, h.w * w);
}

extern "C" void kernel_launch(void* const* d_in, const int* in_sizes, int n_in,
                              void* d_out, int out_size, void* d_ws, size_t ws_size,
                              hipStream_t stream) {
    const float*     x   = (const float*)d_in[0];
    const long long* ei  = (const long long*)d_in[1];   // int64 [2, E]
    const float*     W1  = (const float*)d_in[2];
    const float*     b1  = (const float*)d_in[3];
    const float*     W2  = (const float*)d_in[4];
    const float*     b2  = (const float*)d_in[5];
    const float*     W3  = (const float*)d_in[6];
    const float*     b3  = (const float*)d_in[7];
    (void)in_sizes; (void)n_in; (void)out_size; (void)ws_size;

    const int N = NNODES, E = NEDGES;
    const int F0 = 78,  F1 = 156, F2 = 234, F3 = 312;
    const int F0p = 80, F1p = 160, F2p = 240, F3p = 320;
    const long long* src = ei;
    const long long* dst = ei + E;

    // workspace carve (256B aligned)
    char* ws = (char*)d_ws;
    size_t off = 0;
    auto carve = [&](size_t bytes) -> void* {
        off = (off + 255) & ~(size_t)255;
        void* p = ws + off;
        off += bytes;
        return p;
    };
    float* H    = (float*)carve((size_t)N * F3p * 4);  // GEMM outputs (256 MB)
    float* AGG  = (float*)carve((size_t)N * F2p * 4);  // xp / agg1 / agg2 (192 MB)
    float* deg  = (float*)carve((size_t)N * 4);
    float* dinv = (float*)carve((size_t)N * 4);
    float* nrm  = (float*)carve((size_t)E * 4);
    float* W1t  = (float*)carve((size_t)F1p * F0p * 4);
    float* W2t  = (float*)carve((size_t)F2p * F1p * 4);
    float* W3t  = (float*)carve((size_t)F3p * F2p * 4);
    float* out  = (float*)d_out;

    const int T = 256;
    auto nb = [](long long n, int t) { return (unsigned)((n + t - 1) / t); };

    // degrees + norms (layer-invariant)
    hipMemsetAsync(deg, 0, (size_t)N * 4, stream);
    k_deg <<<nb(E, T), T, 0, stream>>>(dst, deg, E);
    k_dinv<<<nb(N, T), T, 0, stream>>>(deg, dinv, N);
    k_norm<<<nb(E, T), T, 0, stream>>>(src, dst, dinv, nrm, E);

    // pack weights (transposed, zero padded) + input
    k_pack_wt<<<nb((long long)F1p * F0p, T), T, 0, stream>>>(W1, W1t, F0, F0p, F1, F1p);
    k_pack_wt<<<nb((long long)F2p * F1p, T), T, 0, stream>>>(W2, W2t, F1, F1p, F2, F2p);
    k_pack_wt<<<nb((long long)F3p * F2p, T), T, 0, stream>>>(W3, W3t, F2, F2p, F3, F3p);
    k_pack_x <<<nb((long long)N * F0p, T), T, 0, stream>>>(x, AGG, N, F0, F0p);

    const unsigned rowBlocks = N / 64;   // 3125 blocks: 2 waves x 32 rows

    // ---- layer 1: H = xp @ W1 ; AGG = scatter(H) ----
    k_gemm_wmma<<<dim3(rowBlocks, F1p / 80), 64, 0, stream>>>(AGG, W1t, H, F0p, F1p);
    k_init_agg<<<nb((long long)N * (F1p / 4), T), T, 0, stream>>>(H, dinv, b1, AGG, N, F1p, F1, F1p);
    k_scatter <<<nb((long long)E * ((F1 + 3) / 4), T), T, 0, stream>>>(H, src, dst, nrm, AGG, E, F1p, F1, F1p, (F1 + 3) / 4);

    // ---- layer 2 ----
    k_gemm_wmma<<<dim3(rowBlocks, F2p / 80), 64, 0, stream>>>(AGG, W2t, H, F1p, F2p);
    k_init_agg<<<nb((long long)N * (F2p / 4), T), T, 0, stream>>>(H, dinv, b2, AGG, N, F2p, F2, F2p);
    k_scatter <<<nb((long long)E * ((F2 + 3) / 4), T), T, 0, stream>>>(H, src, dst, nrm, AGG, E, F2p, F2, F2p, (F2 + 3) / 4);

    // ---- layer 3: scatter straight into d_out (stride F3 = 312) ----
    k_gemm_wmma<<<dim3(rowBlocks, F3p / 80), 64, 0, stream>>>(AGG, W3t, H, F2p, F3p);
    k_init_agg<<<nb((long long)N * (F3p / 4), T), T, 0, stream>>>(H, dinv, b3, out, N, F3p, F3, F3);
    k_scatter <<<nb((long long)E * ((F3 + 3) / 4), T), T, 0, stream>>>(H, src, dst, nrm, out, E, F3p, F3, F3, (F3 + 3) / 4);
}